// GNN_GBNeck_43138651521675
// MI455X (gfx1250) — compile-verified
//
#include <hip/hip_runtime.h>
#include <hip/hip_bf16.h>

// GB-Neck per-atom solvation energy + analytic forces, MI455X (gfx1250).
// One molecule (64 atoms) per 64-thread block (2 wave32s).
// 64x64 pairwise distance matrix via V_WMMA_F32_16X16X4_F32 (exact f32 Gram,
// K=4 = xyz+pad), staged to LDS (stride 65 -> conflict-free), reused by the
// three VALU passes. All divisions/sqrt/log/exp/tanh use the ~1-ulp hardware
// units (v_rcp/v_sqrt/v_rsq/v_log/v_exp) since the kernel is VALU-bound.

#define NA 64
#define LDS_STRIDE 65

#define RADIUS_C 0.4f
#define OFFSET_C 0.009f
#define ALPHA_C 1.0f
#define BETA_C 0.8f
#define GAMMA_C 4.85f
#define PREF_C (-138.935485f * (1.0f - 1.0f / 78.5f))

typedef __attribute__((ext_vector_type(2))) float v2f;
typedef __attribute__((ext_vector_type(8))) float v8f;

__device__ __forceinline__ float frcp(float x)  { return __builtin_amdgcn_rcpf(x); }
__device__ __forceinline__ float fsqrt(float x) { return __builtin_amdgcn_sqrtf(x); }
__device__ __forceinline__ float frsq(float x)  { return __builtin_amdgcn_rsqf(x); }

// tanh via exp: saturates cleanly to +/-1 for large |t| (2/inf -> 0).
__device__ __forceinline__ float ftanh(float t) {
    float ex = __expf(2.0f * t);
    return 1.0f - 2.0f * frcp(ex + 1.0f);
}

// dI/dd for the GB integral, gated by the valid condition (rho < U).
// L = max(|d - s|, rho): subgradient matches JAX's maximum/abs composition.
__device__ __forceinline__ float dI_dd(float d, float s, float r, float invd) {
    float U = d + s;
    if (r >= U) return 0.0f;                 // not valid -> no gradient flows
    float ads = fabsf(d - s);
    float L   = fmaxf(ads, r);
    float Lp  = (ads > r) ? ((d >= s) ? 1.0f : -1.0f) : 0.0f;
    float invU  = frcp(U);
    float invL  = frcp(L);
    float invU2 = invU * invU;
    float invL2 = invL * invL;
    float A   = 0.25f * (d - s * s * invd);
    float Ap  = 0.25f * (1.0f + s * s * invd * invd);
    float Bt  = invU2 - invL2;
    float Btp = 2.0f * (Lp * invL2 * invL - invU2 * invU);
    float Cp  = 0.5f * ((Lp * invL - invU) * invd - __logf(L * invU) * invd * invd);
    return 0.5f * (invU2 - Lp * invL2 + Ap * Bt + A * Btp + Cp);
}

__global__ __launch_bounds__(NA) void gbneck_kernel(
    const float* __restrict__ pos,   // [B*64, 3]
    const float* __restrict__ gbp,   // [64, 3] (charge, radius, scale)
    float* __restrict__ energy,      // [B]
    float* __restrict__ forces)      // [B*64, 3]
{
    // coordinate planes: sP[0..63]=x, [64..127]=y, [128..191]=z, [192..255]=0
    __shared__ float sP[4 * NA];
    __shared__ float sN[NA];
    __shared__ float sQ[NA], sRho[NA], sSr[NA];
    __shared__ float sBorn[NA], sG[NA], sE[NA];
    __shared__ float sD[NA * LDS_STRIDE];

    const int m = blockIdx.x;
    const int i = threadIdx.x;           // atom index within molecule
    const int g = m * NA + i;            // global atom index

    // ---- load atom data (center coords to kill Gram cancellation; diffs unchanged)
    float px = pos[g * 3 + 0] - 0.6f;
    float py = pos[g * 3 + 1] - 0.6f;
    float pz = pos[g * 3 + 2] - 0.6f;
    float q   = gbp[i * 3 + 0];
    float rad = gbp[i * 3 + 1];
    float sc  = gbp[i * 3 + 2];
    float rho = rad - OFFSET_C;
    float sr  = sc * rho;

    sP[i]          = px;
    sP[NA + i]     = py;
    sP[2 * NA + i] = pz;
    sP[3 * NA + i] = 0.0f;               // K=3 pad plane (branchless operand fetch)
    sN[i]  = px * px + py * py + pz * pz;
    sQ[i] = q; sRho[i] = rho; sSr[i] = sr;
    __syncthreads();

    // ---- WMMA stage: 64x64 distance matrix via f32 Gram (16 tiles of 16x16)
    // A (16x4): lanes 0-15 -> (K0,K1)=(x,y); lanes 16-31 -> (K2,K3)=(z,0)
    // B (4x16): same split, N = lane%16.  Branchless via plane offset half*128.
    {
        const int lane = i & 31;
        const int wave = i >> 5;
        const int half = lane >> 4;
        const int l16  = lane & 15;
        const int cofs = half * 2 * NA;  // 0 -> x/y planes, 128 -> z/pad planes
        for (int t = 0; t < 8; ++t) {
            int ti = wave * 2 + (t >> 2);
            int tj = t & 3;
            int row = ti * 16 + l16;
            int col = tj * 16 + l16;
            v2f a, b;
            a.x = sP[cofs + row];
            a.y = sP[cofs + NA + row];
            b.x = sP[cofs + col];
            b.y = sP[cofs + NA + col];
            v8f c = {};
            c = __builtin_amdgcn_wmma_f32_16x16x4_f32(
                    false, a, false, b, (short)0, c, false, false);
            // Branchless writeback: always sqrt+store; diagonal fixed after
            // a barrier below (avoids 8 exec-mask branch pairs per tile).
            float nN = sN[col];              // invariant over v
            #pragma unroll
            for (int v = 0; v < 8; ++v) {
                int M = ti * 16 + v + half * 8;
                float d2v = sN[M] + nN - 2.0f * c[v];
                sD[M * LDS_STRIDE + tj * 16 + l16] = fsqrt(fmaxf(d2v, 0.0f));
            }
        }
    }
    __syncthreads();
    // diagonal := 1.0 (> RADIUS) so the passes need no j==i test and no rcp(0)
    sD[i * LDS_STRIDE + i] = 1.0f;
    __syncthreads();

    // ---- Pass 1: Isum_i -> born_i, chain = d(born)/d(Isum)
    float Isum = 0.0f;
    for (int j = 0; j < NA; ++j) {
        float d = sD[i * LDS_STRIDE + j];
        if (d < RADIUS_C) {
            float s = sSr[j];
            float U = d + s;
            if (rho < U) {
                float L = fmaxf(fabsf(d - s), rho);
                float invU = frcp(U);
                float invL = frcp(L);
                float invd = frcp(d);
                float term = invL - invU
                           + 0.25f * (d - s * s * invd) * (invU * invU - invL * invL)
                           + 0.5f * __logf(L * invU) * invd;
                Isum += 0.5f * term;
            }
        }
    }
    float psi  = Isum * rho;
    float tArg = psi * (ALPHA_C + psi * (-BETA_C + GAMMA_C * psi));
    float th   = ftanh(tArg);
    float invRad = frcp(rad);
    float born = frcp(frcp(rho) - th * invRad);
    float dtdp = ALPHA_C + psi * (-2.0f * BETA_C + 3.0f * GAMMA_C * psi);
    float chain = born * born * (1.0f - th * th) * invRad * dtdp * rho; // dborn/dIsum
    sBorn[i] = born;
    __syncthreads();

    // ---- Pass 2: pair energy, direct force, dE/dborn_i -> G_i = dE/dIsum_i
    float e = 0.0f, dEdB = 0.0f;
    float fx = 0.0f, fy = 0.0f, fz = 0.0f;
    for (int j = 0; j < NA; ++j) {
        float d = sD[i * LDS_STRIDE + j];
        if (d < RADIUS_C) {
            float d2 = d * d;
            float Bj = sBorn[j];
            float u  = born * Bj;
            float g4 = 0.25f * d2 * frcp(u);       // d2/(4 Bi Bj)
            float ge = __expf(-g4);
            float fgb2 = d2 + u * ge;
            float invf = frsq(fgb2);               // 1/fgb (fgb itself unneeded)
            float invf3 = invf * invf * invf;
            float qq = q * sQ[j];
            e += 0.5f * PREF_C * qq * invf;
            // dE/dBi += PREF*qq*(-1/fgb^2) * Bj*ge*(1+g4)/(2*fgb)
            dEdB -= PREF_C * qq * Bj * ge * (1.0f + g4) * 0.5f * invf3;
            // direct force: F_i += PREF*qq*(1 - ge/4)/fgb^3 * (p_i - p_j)
            float cdir = PREF_C * qq * (1.0f - 0.25f * ge) * invf3;
            fx += cdir * (px - sP[j]);
            fy += cdir * (py - sP[NA + j]);
            fz += cdir * (pz - sP[2 * NA + j]);
        }
    }
    float invB = frcp(born);
    e    += 0.5f * PREF_C * q * q * invB;               // self energy
    dEdB -= 0.5f * PREF_C * q * q * invB * invB;        // self d/dborn
    sG[i] = dEdB * chain;   // dE/dIsum_i
    sE[i] = e;
    __syncthreads();

    // ---- per-molecule energy (thread 0; sE is read-only from here on)
    if (i == 0) {
        float es = 0.0f;
        for (int k = 0; k < NA; ++k) es += sE[k];
        energy[m] = es;
    }

    // ---- Pass 3: Born-mediated forces
    float Gi = sG[i];
    for (int j = 0; j < NA; ++j) {
        float d = sD[i * LDS_STRIDE + j];
        if (d < RADIUS_C) {
            float invd = frcp(d);
            float dij = dI_dd(d, sSr[j], rho, invd);   // dI_ij/dd  (rho_i, sr_j)
            float dji = dI_dd(d, sr, sRho[j], invd);   // dI_ji/dd  (rho_j, sr_i)
            float coeff = (Gi * dij + sG[j] * dji) * invd;
            fx -= coeff * (px - sP[j]);
            fy -= coeff * (py - sP[NA + j]);
            fz -= coeff * (pz - sP[2 * NA + j]);
        }
    }

    forces[g * 3 + 0] = fx;
    forces[g * 3 + 1] = fy;
    forces[g * 3 + 2] = fz;
}

extern "C" void kernel_launch(void* const* d_in, const int* in_sizes, int n_in,
                              void* d_out, int out_size, void* d_ws, size_t ws_size,
                              hipStream_t stream) {
    const float* pos = (const float*)d_in[0];   // [B*64, 3] f32
    const float* gbp = (const float*)d_in[1];   // [64, 3]   f32
    // d_in[2] = batch indices (implicit: molecule = blockIdx.x), unused
    int n = in_sizes[1] / 3;                    // 64 atoms/molecule
    int B = in_sizes[0] / (3 * n);              // 4096 molecules
    float* energy = (float*)d_out;              // [B,1] first
    float* forces = (float*)d_out + B;          // then [B*n,3]
    gbneck_kernel<<<B, NA, 0, stream>>>(pos, gbp, energy, forces);
}